// SuperLoss_27075473834381
// MI455X (gfx1250) — compile-verified
//
#include <hip/hip_runtime.h>
#include <hip/hip_bf16.h>
#include <stdint.h>

// SuperLoss elementwise kernel for MI455X (gfx1250).
// loss = sigma*L + W^2, where L = BCE-with-logits(pr, gt),
// W = LambertW(0.5*max(BETA0, L)) via 8 Halley iterations,
// sigma = exp(-W)  (so log(sigma) == -W exactly).

#define TPB 256
#define SL_BETA0 (-2.0f / (2.7182818284590452f + 0.08f))

#if defined(__gfx1250__) && __has_builtin(__builtin_amdgcn_global_load_async_to_lds_b128)
#define HAVE_ASYNC_LDS 1
#else
#define HAVE_ASYNC_LDS 0
#endif

#if HAVE_ASYNC_LDS
// Builtin signature (from clang-22 diagnostic): param0 is
// 'int __attribute__((vector_size(16))) __device__ *' (AS1 v4i*).
typedef int sl_v4i __attribute__((vector_size(4 * sizeof(int))));
typedef __attribute__((address_space(1))) sl_v4i as1_v4i;
typedef __attribute__((address_space(3))) sl_v4i as3_v4i;

__device__ __forceinline__ as1_v4i* sl_as_global(const void* p) {
  // Generic device pointers to global memory are numerically the global address.
  return (as1_v4i*)(unsigned long long)(uintptr_t)p;
}
__device__ __forceinline__ as3_v4i* sl_as_lds(const void* p) {
  // HW truncates generic LDS addresses to addr[31:0] (ISA 10.2); mimic that.
  return (as3_v4i*)(unsigned int)(uintptr_t)p;
}
#endif

__device__ __forceinline__ void sl_wait_asynccnt_le(int n) {
#if defined(__gfx1250__)
  if (n == 0) asm volatile("s_wait_asynccnt 0x0" ::: "memory");
  else        asm volatile("s_wait_asynccnt 0x2" ::: "memory");
#endif
}

__device__ __forceinline__ float sl_lambertw(float y) {
  // Principal branch, y >= 0.5*BETA0 > -1/e. Seed: log1p(y).
  float w = __logf(1.0f + y);
#pragma unroll
  for (int i = 0; i < 8; ++i) {
    float e = __expf(w);                       // v_exp_f32
    float f = __fmaf_rn(w, e, -y);             // w*e - y
    float t = w + 1.0f;
    // Halley: w -= f / (e*t - (w+2)*f/(2t))  ==  w -= 2t*f / (2t^2*e - (w+2)*f)
    float num = 2.0f * t * f;
    float den = __fmaf_rn(2.0f * t * t, e, -((w + 2.0f) * f));
    w = w - __fdividef(num, den);              // v_rcp_f32 + mul
  }
  return w;
}

__device__ __forceinline__ float sl_elem(float pr, float gt) {
  // Stable logaddexp(0, pr) = max(pr,0) + log1p(exp(-|pr|))
  float L = fmaxf(pr, 0.0f) - pr * gt + __logf(1.0f + __expf(-fabsf(pr)));
  float y = 0.5f * fmaxf(SL_BETA0, L);
  float w = sl_lambertw(y);
  float sigma = __expf(-w);
  // LAM=1, TAU=0; log(sigma) = -w  =>  loss = sigma*L + w^2
  return __fmaf_rn(sigma, L, w * w);
}

__device__ __forceinline__ float4 sl_vec4(float4 a, float4 b) {
  float4 r;
  r.x = sl_elem(a.x, b.x);
  r.y = sl_elem(a.y, b.y);
  r.z = sl_elem(a.z, b.z);
  r.w = sl_elem(a.w, b.w);
  return r;
}

__global__ __launch_bounds__(TPB) void superloss_kernel(
    const float* __restrict__ prf, const float* __restrict__ gtf,
    float* __restrict__ outf, long long n) {
  const int tid = (int)threadIdx.x;
  const long long n4 = n >> 2;
  const long long tiles = n4 / TPB;
  const long long bstart = (long long)blockIdx.x;
  const long long bstep = (long long)gridDim.x;

  const float4* __restrict__ pr4 = (const float4*)prf;
  const float4* __restrict__ gt4 = (const float4*)gtf;
  float4* __restrict__ out4 = (float4*)outf;

#if HAVE_ASYNC_LDS
  // Depth-2 async global->LDS pipeline. Each lane stages & consumes its own
  // 16B slot, so per-wave ASYNCcnt ordering is sufficient (no barriers).
  __shared__ float4 sPr[2][TPB];
  __shared__ float4 sGt[2][TPB];

  long long t = bstart;
  if (t < tiles) {
    __builtin_amdgcn_global_load_async_to_lds_b128(
        sl_as_global(pr4 + t * TPB + tid), sl_as_lds(&sPr[0][tid]), 0, 0);
    __builtin_amdgcn_global_load_async_to_lds_b128(
        sl_as_global(gt4 + t * TPB + tid), sl_as_lds(&sGt[0][tid]), 0, 0);
  }
  int buf = 0;
  while (t < tiles) {
    const long long tn = t + bstep;
    if (tn < tiles) {
      __builtin_amdgcn_global_load_async_to_lds_b128(
          sl_as_global(pr4 + tn * TPB + tid), sl_as_lds(&sPr[buf ^ 1][tid]), 0, 0);
      __builtin_amdgcn_global_load_async_to_lds_b128(
          sl_as_global(gt4 + tn * TPB + tid), sl_as_lds(&sGt[buf ^ 1][tid]), 0, 0);
      sl_wait_asynccnt_le(2);   // retire current stage, keep next in flight
    } else {
      sl_wait_asynccnt_le(0);
    }
    const float4 a = sPr[buf][tid];   // ds_load_b128
    const float4 b = sGt[buf][tid];
    out4[t * TPB + tid] = sl_vec4(a, b);
    t = tn;
    buf ^= 1;
  }
  // Leftover float4s past the last full tile.
  for (long long i = tiles * TPB + bstart * TPB + tid; i < n4; i += bstep * TPB) {
    out4[i] = sl_vec4(pr4[i], gt4[i]);
  }
#else
  // Fallback: direct b128 streaming loads + gfx1250 prefetch of next stride.
  const long long stride = bstep * TPB;
  for (long long i = bstart * TPB + tid; i < n4; i += stride) {
    if (i + stride < n4) {
      __builtin_prefetch(pr4 + i + stride, 0, 0);   // global_prefetch_b8
      __builtin_prefetch(gt4 + i + stride, 0, 0);
    }
    out4[i] = sl_vec4(pr4[i], gt4[i]);
  }
#endif

  // Scalar tail (n not divisible by 4) — block 0 only.
  if (blockIdx.x == 0) {
    for (long long i = (n4 << 2) + tid; i < n; i += TPB) {
      outf[i] = sl_elem(prf[i], gtf[i]);
    }
  }
}

extern "C" void kernel_launch(void* const* d_in, const int* in_sizes, int n_in,
                              void* d_out, int out_size, void* d_ws, size_t ws_size,
                              hipStream_t stream) {
  (void)n_in; (void)out_size; (void)d_ws; (void)ws_size;
  const float* pr = (const float*)d_in[0];
  const float* gt = (const float*)d_in[1];
  float* out = (float*)d_out;
  const long long n = (long long)in_sizes[0];

  const long long n4 = n >> 2;
  const long long tiles = n4 / TPB;
  long long want = tiles > 0 ? tiles : 1;
  if (want > 2048) want = 2048;   // persistent blocks; ~16 tiles each at full size
  superloss_kernel<<<(int)want, TPB, 0, stream>>>(pr, gt, out, n);
}